// CrossAttentionBlock_42185168782056
// MI455X (gfx1250) — compile-verified
//
#include <hip/hip_runtime.h>

// ---------------------------------------------------------------------------
// CrossAttentionBlock, exact-math simplified:
//   context length == 1  =>  softmax == 1 exactly  =>  out = x + bcast(y)
//   v[b] = Wkv[256:,:] @ context[b] + bkv[256:]     (K-half / GroupNorm / Wq dead)
//   y[b] = Wout @ v[b] + bout
//   out[b,c,h,w] = x[b,c,h,w] + y[b,c]
// ---------------------------------------------------------------------------

typedef __attribute__((ext_vector_type(2))) float v2f;
typedef __attribute__((ext_vector_type(4))) float v4f;
typedef __attribute__((ext_vector_type(8))) float v8f;

#define BATCH 32
#define CCH   256
#define CTXD  512
#define HWSP  4096   // 64*64

// ---------------------------------------------------------------------------
// Fused projection kernel: one workgroup, 32 waves, wave-per-16x16-tile.
// Stage A: V[32,256] = context[32,512] @ Wkv_v^T[512,256] + bkv_v   -> LDS
// Stage B: Y[32,256] = V[32,256] @ Wout^T[256,256] + bout           -> global ws
//
// WMMA f32 16x16x4 per-lane layout (ISA 7.12.2, wave32):
//   A (16x4, MxK):  lane l holds rows m=l&15; VGPR0/1 = K = 2*(l>>4)+{0,1}
//   B (4x16, KxN):  lane l holds col  n=l&15; VGPR0/1 = K = 2*(l>>4)+{0,1}
//   C/D (16x16):    VGPR r, lane l -> M = r + 8*(l>>4), N = l&15
// ---------------------------------------------------------------------------
__global__ __launch_bounds__(1024) void
cab_proj_wmma_kernel(const float* __restrict__ ctx,   // [32,512]
                     const float* __restrict__ Wkv,   // [512,512] row-major
                     const float* __restrict__ bkv,   // [512]
                     const float* __restrict__ Wout,  // [256,256] row-major
                     const float* __restrict__ bout,  // [256]
                     float* __restrict__ Y)           // [32,256]
{
    __shared__ float Vsh[BATCH * CCH];   // 32 KB

    const int lane  = threadIdx.x & 31;
    const int wave  = threadIdx.x >> 5;  // 0..31
    const int mtile = wave >> 4;         // 0..1  (M = 32 -> two 16-row tiles)
    const int ntile = wave & 15;         // 0..15 (N = 256 -> sixteen tiles)

    const int lrow  = lane & 15;         // M (for A) / N (for B) within tile
    const int khalf = lane >> 4;         // selects K-pair {0,1} vs {2,3}

    const int mg = mtile * 16 + lrow;    // global batch row   0..31
    const int ng = ntile * 16 + lrow;    // global out channel 0..255

    // ---------------- Stage A: V = context @ Wkv_v^T + bkv_v ----------------
    v8f acc = {0.f, 0.f, 0.f, 0.f, 0.f, 0.f, 0.f, 0.f};
    {
        const float* arow = ctx + mg * CTXD;              // context[mg, :]
        const float* brow = Wkv + (CCH + ng) * CTXD;      // Wkv_v[ng, :]
        for (int k = 0; k < CTXD; k += 4) {
            const int kb = k + khalf * 2;
            v2f a, b;
            a.x = arow[kb]; a.y = arow[kb + 1];
            b.x = brow[kb]; b.y = brow[kb + 1];
            acc = __builtin_amdgcn_wmma_f32_16x16x4_f32(
                false, a, false, b, (short)0, acc, false, false);
        }
        const float bias = bkv[CCH + ng];
#pragma unroll
        for (int r = 0; r < 8; ++r) {
            const int mo = mtile * 16 + r + 8 * khalf;    // C/D layout
            Vsh[mo * CCH + ng] = acc[r] + bias;
        }
    }
    __syncthreads();

    // ---------------- Stage B: Y = V @ Wout^T + bout ------------------------
    v8f acc2 = {0.f, 0.f, 0.f, 0.f, 0.f, 0.f, 0.f, 0.f};
    {
        const float* arow = Vsh + mg * CCH;               // V[mg, :] (LDS)
        const float* brow = Wout + ng * CCH;              // Wout[ng, :]
        for (int k = 0; k < CCH; k += 4) {
            const int kb = k + khalf * 2;
            v2f a, b;
            a.x = arow[kb]; a.y = arow[kb + 1];
            b.x = brow[kb]; b.y = brow[kb + 1];
            acc2 = __builtin_amdgcn_wmma_f32_16x16x4_f32(
                false, a, false, b, (short)0, acc2, false, false);
        }
        const float bias = bout[ng];
#pragma unroll
        for (int r = 0; r < 8; ++r) {
            const int mo = mtile * 16 + r + 8 * khalf;
            Y[mo * CCH + ng] = acc2[r] + bias;
        }
    }
}

// ---------------------------------------------------------------------------
// Streaming broadcast add: out = x + Y[b,c], float4 per thread.
// 268 MB total traffic -> HBM-bandwidth bound (~11.5 us at 23.3 TB/s).
// Nontemporal hints keep the one-shot stream out of cache (Y stays cached).
// ---------------------------------------------------------------------------
__global__ __launch_bounds__(256) void
cab_add_bcast_kernel(const v4f* __restrict__ x4,
                     const float* __restrict__ Y,    // [32*256]
                     v4f* __restrict__ out4,
                     long long n4)
{
    const long long i = (long long)blockIdx.x * blockDim.x + threadIdx.x;
    if (i >= n4) return;
    const int row = (int)(i >> 10);            // HWSP/4 = 1024 float4 per (b,c)
    const float y = Y[row];
    v4f v = __builtin_nontemporal_load(&x4[i]);
    v.x += y; v.y += y; v.z += y; v.w += y;
    __builtin_nontemporal_store(v, &out4[i]);
}

extern "C" void kernel_launch(void* const* d_in, const int* in_sizes, int n_in,
                              void* d_out, int out_size, void* d_ws, size_t ws_size,
                              hipStream_t stream) {
    // setup_inputs order: x, context, gn_w, gn_b, Wq, bq, Wkv, bkv, Wout, bout
    const float* x    = (const float*)d_in[0];
    const float* ctx  = (const float*)d_in[1];
    const float* Wkv  = (const float*)d_in[6];
    const float* bkv  = (const float*)d_in[7];
    const float* Wout = (const float*)d_in[8];
    const float* bout = (const float*)d_in[9];
    float* out = (float*)d_out;
    float* Y   = (float*)d_ws;     // 32*256 floats = 32 KB scratch

    // Tiny fused WMMA projection (single workgroup, 32 waves).
    cab_proj_wmma_kernel<<<1, 1024, 0, stream>>>(ctx, Wkv, bkv, Wout, bout, Y);

    // Bandwidth-bound broadcast add over 32*256*64*64 elements.
    const long long n4 = (long long)out_size / 4;   // 8,388,608 float4
    const int threads = 256;
    const unsigned blocks = (unsigned)((n4 + threads - 1) / threads);
    cab_add_bcast_kernel<<<blocks, threads, 0, stream>>>(
        (const v4f*)x, Y, (v4f*)out, n4);
}